// E_Attention_11441792876628
// MI455X (gfx1250) — compile-verified
//
#include <hip/hip_runtime.h>

// Problem constants (from reference): B=8, S=4096, D=256, fp32 everywhere.
#define BATCH 8
#define SEQ   4096
#define DIM   256
#define NEGV  (-1000000000.0f)
#define PAD   72      // LDS row stride (floats): rows r and r+2 differ by 144
                      // words = +16 banks mod 64 -> half-waves hit disjoint banks

typedef __attribute__((ext_vector_type(2))) float v2f;
typedef __attribute__((ext_vector_type(8))) float v8f;

// D = A(16x4 f32) * B(4x16 f32) + C(16x16 f32)  -> v_wmma_f32_16x16x4_f32
__device__ __forceinline__ v8f wmma4_f32(v2f a, v2f b, v8f c) {
  return __builtin_amdgcn_wmma_f32_16x16x4_f32(
      /*neg_a=*/false, a, /*neg_b=*/false, b,
      /*c_mod=*/(short)0, c, /*reuse_a=*/false, /*reuse_b=*/false);
}

// Async global->LDS 128-bit copy (ASYNCcnt path). ldsAddr = LDS byte offset
// (low 32 bits of the generic pointer per the flat-aperture rule).
__device__ __forceinline__ void async_copy_b128(unsigned ldsAddr, const float* gp) {
  asm volatile("global_load_async_to_lds_b128 %0, %1, off"
               :: "v"(ldsAddr), "v"(gp) : "memory");
}
__device__ __forceinline__ void wait_async0() {
  asm volatile("s_wait_asynccnt 0x0" ::: "memory");
}

// ---------------------------------------------------------------------------
// Kernel 0: zero the s1 accumulation buffer (split-K uses atomics into it).
// ---------------------------------------------------------------------------
__global__ void zero_kernel(float* __restrict__ p, int n4) {
  int i = blockIdx.x * blockDim.x + threadIdx.x;
  if (i < n4) {
    float4 z = make_float4(0.f, 0.f, 0.f, 0.f);
    reinterpret_cast<float4*>(p)[i] = z;
  }
}

// ---------------------------------------------------------------------------
// Kernel 1: per-row stats. One wave32 per row (b,s):
//   scale = mask ? 0 : 1/sum(q^2);  bias = mask ? NEG : 0;  cb = 1/sum(k^2)
// ---------------------------------------------------------------------------
__global__ __launch_bounds__(256)
void stats_kernel(const float* __restrict__ q, const float* __restrict__ k,
                  const unsigned char* __restrict__ mask,
                  float* __restrict__ scale, float* __restrict__ bias,
                  float* __restrict__ cb) {
  const int lane = threadIdx.x & 31;
  const int row  = blockIdx.x * 8 + (threadIdx.x >> 5);   // 8 waves / block
  const float* qr = q + (size_t)row * DIM;
  const float* kr = k + (size_t)row * DIM;

  float sq = 0.f, sk = 0.f;
#pragma unroll
  for (int j = 0; j < 2; ++j) {           // 32 lanes * 2 * float4 = 256 elems
    const float4 vq = *reinterpret_cast<const float4*>(qr + lane * 4 + j * 128);
    const float4 vk = *reinterpret_cast<const float4*>(kr + lane * 4 + j * 128);
    sq += vq.x * vq.x + vq.y * vq.y + vq.z * vq.z + vq.w * vq.w;
    sk += vk.x * vk.x + vk.y * vk.y + vk.z * vk.z + vk.w * vk.w;
  }
#pragma unroll
  for (int off = 16; off > 0; off >>= 1) {  // wave32 butterfly reduction
    sq += __shfl_xor(sq, off, 32);
    sk += __shfl_xor(sk, off, 32);
  }
  if (lane == 0) {
    const bool m = mask[row] != 0;
    scale[row] = m ? 0.0f : 1.0f / sq;
    bias[row]  = m ? NEGV : 0.0f;
    cb[row]    = 1.0f / sk;
  }
}

// ---------------------------------------------------------------------------
// Kernel 2: s1[b,d,e] += sum_{s in chunk} (K[b,s,d]/b_s) * V[b,s,e]
// grid = (e-strips=4, k-chunks=8, batch=8), block = 512 threads = 16 waves
// (one per 16-row d tile). The 512x64 V slice is streamed through LDS in
// 64-row sub-chunks with double-buffered async global->LDS copies; WMMA B
// operands come from LDS. Split-K combined with global_atomic_add_f32.
// ---------------------------------------------------------------------------
__global__ __launch_bounds__(512)
void phase1_kernel(const float* __restrict__ K, const float* __restrict__ V,
                   const float* __restrict__ cb, float* __restrict__ s1) {
  __shared__ float vs[2][64 * PAD];        // 2 x 18 KB double buffer

  const int tid  = threadIdx.x;
  const int lane = tid & 31;
  const int wave = tid >> 5;               // 0..15 -> d tile
  const int half = lane >> 4;              // lanes 16-31 handle K+2,K+3
  const int lm   = lane & 15;
  const int klo  = half << 1;

  const int e0   = blockIdx.x * 64;
  const int sBeg = blockIdx.y * (SEQ / 8);
  const int b    = blockIdx.z;
  const int d0   = wave * 16;

  const size_t bSD = (size_t)b * SEQ * DIM;
  const float* Kb  = K + bSD;
  const float* Vb  = V + bSD + e0;
  const float* cbb = cb + (size_t)b * SEQ;

  // stage one 64x64 V sub-chunk (1024 x b128) into LDS buffer `buf`
  auto stage = [&](int buf, int sBase) {
#pragma unroll
    for (int i = 0; i < 2; ++i) {
      const int c   = tid + i * 512;       // 4-float chunk id, 0..1023
      const int row = c >> 4;              // 16 chunks per 64-col row
      const int col = (c & 15) << 2;
      const unsigned lo = (unsigned)(size_t)&vs[buf][row * PAD + col];
      async_copy_b128(lo, Vb + (size_t)(sBase + row) * DIM + col);
    }
  };

  stage(0, sBeg);
  wait_async0();
  __syncthreads();

  v8f acc0 = {}, acc1 = {}, acc2 = {}, acc3 = {};
  int buf = 0;
  for (int sub = 0; sub < 8; ++sub) {
    const int sBase = sBeg + sub * 64;
    if (sub + 1 < 8) stage(buf ^ 1, sBase + 64);   // overlap copy with math

    for (int s = 0; s < 64; s += 4) {
      const int r0 = sBase + s + klo;
      const float c0 = cbb[r0];
      const float c1 = cbb[r0 + 1];
      const float* k0 = Kb + (size_t)r0 * DIM + d0 + lm;
      __builtin_prefetch(k0 + 8 * DIM, 0, 0);      // global_prefetch_b8

      v2f a;                                       // A[m,k] = K[s+k,d0+m]/b_s
      a.x = k0[0]   * c0;
      a.y = k0[DIM] * c1;

      const int p0 = (s + klo) * PAD;
      const int p1 = p0 + PAD;
      v2f bb;                                      // B[k,n] from LDS
      bb.x = vs[buf][p0 + lm];      bb.y = vs[buf][p1 + lm];      acc0 = wmma4_f32(a, bb, acc0);
      bb.x = vs[buf][p0 + 16 + lm]; bb.y = vs[buf][p1 + 16 + lm]; acc1 = wmma4_f32(a, bb, acc1);
      bb.x = vs[buf][p0 + 32 + lm]; bb.y = vs[buf][p1 + 32 + lm]; acc2 = wmma4_f32(a, bb, acc2);
      bb.x = vs[buf][p0 + 48 + lm]; bb.y = vs[buf][p1 + 48 + lm]; acc3 = wmma4_f32(a, bb, acc3);
    }

    wait_async0();        // next buffer landed (overlapped with the 16 steps)
    __syncthreads();      // everyone done reading `buf` before it is refilled
    buf ^= 1;
  }

  // C/D layout: VGPR r -> rows (r, r+8) split across lane halves.
  float* out = s1 + (size_t)b * DIM * DIM;
  const int rowBase = d0 + half * 8;
#pragma unroll
  for (int r = 0; r < 8; ++r) {
    float* orow = out + (size_t)(rowBase + r) * DIM + e0 + lm;
    unsafeAtomicAdd(orow + 0,  acc0[r]);
    unsafeAtomicAdd(orow + 16, acc1[r]);
    unsafeAtomicAdd(orow + 32, acc2[r]);
    unsafeAtomicAdd(orow + 48, acc3[r]);
  }
}

// ---------------------------------------------------------------------------
// Kernel 3: out[b,s,e] = (1/D) * sum_d (Q[b,s,d]*scale_s + bias_s) * s1[b,d,e]
// grid = (e-strips=4, s-blocks=16, batch=8), block = 512 (16 waves, one per
// 16-row s tile). The 256x64 s1 slice is staged into LDS in two 128-row
// halves via async global->LDS copies; WMMA B operands come from LDS.
// ---------------------------------------------------------------------------
__global__ __launch_bounds__(512)
void phase2_kernel(const float* __restrict__ Q, const float* __restrict__ s1,
                   const float* __restrict__ scale, const float* __restrict__ bias,
                   float* __restrict__ out) {
  __shared__ float bs[128 * PAD];          // 36 KB, one 128-row half of s1

  const int tid  = threadIdx.x;
  const int lane = tid & 31;
  const int wave = tid >> 5;
  const int half = lane >> 4;
  const int lm   = lane & 15;
  const int klo  = half << 1;

  const int e0 = blockIdx.x * 64;
  const int s0 = (blockIdx.y * 16 + wave) * 16;
  const int b  = blockIdx.z;

  const size_t bSD = (size_t)b * SEQ * DIM;
  const float* Qb = Q + bSD;
  const float* S1 = s1 + (size_t)b * DIM * DIM + e0;

  const int   sRow = s0 + lm;              // A row owned by this lane
  const float sc   = scale[(size_t)b * SEQ + sRow];
  const float bi   = bias[(size_t)b * SEQ + sRow];
  const float* qrow = Qb + (size_t)sRow * DIM;

  v8f acc0 = {}, acc1 = {}, acc2 = {}, acc3 = {};

  for (int h = 0; h < 2; ++h) {            // two 128-row halves of the K dim
#pragma unroll
    for (int i = 0; i < 4; ++i) {          // 2048 x b128 copies / 512 threads
      const int c   = tid + i * 512;
      const int row = c >> 4;
      const int col = (c & 15) << 2;
      const unsigned lo = (unsigned)(size_t)&bs[row * PAD + col];
      async_copy_b128(lo, S1 + (size_t)(h * 128 + row) * DIM + col);
    }
    wait_async0();
    __syncthreads();

    for (int d = 0; d < 128; d += 4) {
      const float2 qv = *reinterpret_cast<const float2*>(qrow + h * 128 + d + klo);
      v2f a;                               // score = q*scale + bias
      a.x = qv.x * sc + bi;
      a.y = qv.y * sc + bi;

      const int p0 = (d + klo) * PAD;
      const int p1 = p0 + PAD;
      v2f bb;
      bb.x = bs[p0 + lm];      bb.y = bs[p1 + lm];      acc0 = wmma4_f32(a, bb, acc0);
      bb.x = bs[p0 + 16 + lm]; bb.y = bs[p1 + 16 + lm]; acc1 = wmma4_f32(a, bb, acc1);
      bb.x = bs[p0 + 32 + lm]; bb.y = bs[p1 + 32 + lm]; acc2 = wmma4_f32(a, bb, acc2);
      bb.x = bs[p0 + 48 + lm]; bb.y = bs[p1 + 48 + lm]; acc3 = wmma4_f32(a, bb, acc3);
    }
    __syncthreads();                       // all reads done before next stage
  }

  const float invD = 1.0f / (float)DIM;
  float* ob = out + bSD;
  const int rowBase = s0 + half * 8;
#pragma unroll
  for (int r = 0; r < 8; ++r) {
    float* orow = ob + (size_t)(rowBase + r) * DIM + e0 + lm;
    orow[0]  = acc0[r] * invD;
    orow[16] = acc1[r] * invD;
    orow[32] = acc2[r] * invD;
    orow[48] = acc3[r] * invD;
  }
}

// ---------------------------------------------------------------------------
// Launch: inputs in setup_inputs() order: query, key, value, mask(bool u8).
// Workspace layout (floats): scale[32768] | bias[32768] | cb[32768] | s1[524288]
// = 2.49 MB total.
// ---------------------------------------------------------------------------
extern "C" void kernel_launch(void* const* d_in, const int* in_sizes, int n_in,
                              void* d_out, int out_size, void* d_ws, size_t ws_size,
                              hipStream_t stream) {
  const float*         q    = (const float*)d_in[0];
  const float*         k    = (const float*)d_in[1];
  const float*         v    = (const float*)d_in[2];
  const unsigned char* mask = (const unsigned char*)d_in[3];
  float* out = (float*)d_out;

  float* ws    = (float*)d_ws;
  float* scale = ws;                                  // B*S
  float* bias  = ws + (size_t)BATCH * SEQ;            // B*S
  float* cbv   = ws + 2 * (size_t)BATCH * SEQ;        // B*S
  float* s1    = ws + 3 * (size_t)BATCH * SEQ;        // B*D*D

  const int s1Elems = BATCH * DIM * DIM;              // 524288

  zero_kernel <<<(s1Elems / 4 + 255) / 256, 256, 0, stream>>>(s1, s1Elems / 4);
  stats_kernel<<<(BATCH * SEQ) / 8, 256, 0, stream>>>(q, k, mask, scale, bias, cbv);
  phase1_kernel<<<dim3(4, 8, BATCH),  512, 0, stream>>>(k, v, cbv, s1);
  phase2_kernel<<<dim3(4, 16, BATCH), 512, 0, stream>>>(q, s1, scale, bias, out);
}